// GMMConv_fuse_no_spmm_67508295958897
// MI455X (gfx1250) — compile-verified
//
#include <hip/hip_runtime.h>

// GMMConv forward, fused (no materialized SpMM messages).
// Stage 1: node_feat[N,256] = feat[N,256] @ W_fc[256,256]^T   (fp32 WMMA 16x16x4)
// Stage 2: per CSR row: out[n,f] = bias[f] +
//          sum_{e in row} sum_k exp(-0.5*||(pseudo[e]-mu[k])*inv_sigma[k]||^2) * node_feat[colind[e], k*64+f]

typedef __attribute__((ext_vector_type(2))) float v2f;
typedef __attribute__((ext_vector_type(8))) float v8f;

#define NNODE 50000
#define INFEAT 256
#define KKER 4
#define FOUT 64
#define KF 256            // KKER * FOUT, also the GEMM N dimension
#define MTILES (NNODE / 16)   // 3125 (N divisible by 16)
#define NGROUPS 4             // 4 groups of 64 columns each

// ---------------- Stage 1: fp32 WMMA GEMM ----------------
// One wave computes a 16x64 output tile: 4 accumulators, A fragment reused.
__global__ __launch_bounds__(256) void gmm_gemm_kernel(
    const float* __restrict__ feat,     // [N, 256]
    const float* __restrict__ wfc,      // [256, 256]  (K*F rows, IN cols)
    float* __restrict__ nodefeat)       // [N, 256]
{
    const int lane  = threadIdx.x & 31;
    const int wave  = blockIdx.x * (blockDim.x >> 5) + (threadIdx.x >> 5);
    const int mtile = wave >> 2;          // 0..3124
    const int ngrp  = wave & 3;           // 0..3
    if (mtile >= MTILES) return;

    const int m0   = mtile * 16;
    const int n0   = ngrp * 64;
    const int lrow = lane & 15;           // M (for A) / N (for B) within tile
    const int kh   = (lane >> 4) * 2;     // lanes 16-31 hold K=2,3 halves

    // A 16x4 fragment: lane l<16 -> A[l][kk..kk+1], lane l>=16 -> A[l][kk+2..kk+3]
    const float* aptr = feat + (size_t)(m0 + lrow) * INFEAT + kh;
    // B 4x16 fragment: B[k][n] = W_fc[n][k] -> contiguous float2 per lane
    const float* bptr = wfc + (size_t)(n0 + lrow) * INFEAT + kh;

    v8f acc0 = {}, acc1 = {}, acc2 = {}, acc3 = {};

    for (int kk = 0; kk < INFEAT; kk += 4) {
        v2f a  = *(const v2f*)(aptr + kk);
        v2f b0 = *(const v2f*)(bptr + kk);
        v2f b1 = *(const v2f*)(bptr + 16 * INFEAT + kk);
        v2f b2 = *(const v2f*)(bptr + 32 * INFEAT + kk);
        v2f b3 = *(const v2f*)(bptr + 48 * INFEAT + kk);
        acc0 = __builtin_amdgcn_wmma_f32_16x16x4_f32(false, a, false, b0, (short)0, acc0, false, false);
        acc1 = __builtin_amdgcn_wmma_f32_16x16x4_f32(false, a, false, b1, (short)0, acc1, false, false);
        acc2 = __builtin_amdgcn_wmma_f32_16x16x4_f32(false, a, false, b2, (short)0, acc2, false, false);
        acc3 = __builtin_amdgcn_wmma_f32_16x16x4_f32(false, a, false, b3, (short)0, acc3, false, false);
    }

    // C/D layout: VGPR j, lanes 0-15 -> row j, cols 0-15; lanes 16-31 -> row j+8
    const int rowoff = (lane >> 4) * 8;
    float* dbase = nodefeat + (size_t)(m0 + rowoff) * KF + n0 + lrow;
#pragma unroll
    for (int j = 0; j < 8; ++j) {
        dbase[(size_t)j * KF +  0] = acc0[j];
        dbase[(size_t)j * KF + 16] = acc1[j];
        dbase[(size_t)j * KF + 32] = acc2[j];
        dbase[(size_t)j * KF + 48] = acc3[j];
    }
}

// ---------------- Stage 2: CSR row-wise gather + mixture-weighted reduce ----------------
// One wave per destination row; lane owns output features {2*lane, 2*lane+1}.
__global__ __launch_bounds__(256) void gmm_edge_kernel(
    const int*   __restrict__ rowptr,    // [N+1]
    const int*   __restrict__ colind,    // [E]
    const float* __restrict__ pseudo,    // [E, 2]
    const float* __restrict__ nodefeat,  // [N, 256]
    const float* __restrict__ mu,        // [4, 2]
    const float* __restrict__ invs,      // [4, 2]
    const float* __restrict__ bias,      // [64]
    float* __restrict__ out)             // [N, 64]
{
    const int lane = threadIdx.x & 31;
    const int row  = blockIdx.x * (blockDim.x >> 5) + (threadIdx.x >> 5);
    if (row >= NNODE) return;

    // Kernel parameters (wave-uniform, tiny)
    float mu0[KKER], mu1[KKER], s0[KKER], s1[KKER];
#pragma unroll
    for (int k = 0; k < KKER; ++k) {
        mu0[k] = mu[k * 2 + 0];
        mu1[k] = mu[k * 2 + 1];
        float a = invs[k * 2 + 0], b = invs[k * 2 + 1];
        s0[k] = a * a;
        s1[k] = b * b;
    }

    const int e0 = rowptr[row];
    const int e1 = rowptr[row + 1];

    float acc0 = 0.0f, acc1 = 0.0f;
    for (int e = e0; e < e1; ++e) {
        const int   c  = colind[e];
        const float p0 = pseudo[(size_t)e * 2 + 0];
        const float p1 = pseudo[(size_t)e * 2 + 1];
        const float* src = nodefeat + (size_t)c * KF + lane * 2;
#pragma unroll
        for (int k = 0; k < KKER; ++k) {
            float d0 = p0 - mu0[k];
            float d1 = p1 - mu1[k];
            float g  = __expf(-0.5f * (d0 * d0 * s0[k] + d1 * d1 * s1[k]));
            v2f v = *(const v2f*)(src + k * FOUT);
            acc0 += g * v.x;
            acc1 += g * v.y;
        }
    }

    v2f bo = *(const v2f*)(bias + lane * 2);
    v2f r;
    r.x = acc0 + bo.x;
    r.y = acc1 + bo.y;
    *(v2f*)(out + (size_t)row * FOUT + lane * 2) = r;
}

extern "C" void kernel_launch(void* const* d_in, const int* in_sizes, int n_in,
                              void* d_out, int out_size, void* d_ws, size_t ws_size,
                              hipStream_t stream) {
    // setup_inputs order: rowptr, colind, colptr, rowind, permute, feat, pseudo,
    //                     W_fc, mu, inv_sigma, bias
    const int*   rowptr = (const int*)  d_in[0];
    const int*   colind = (const int*)  d_in[1];
    const float* feat   = (const float*)d_in[5];
    const float* pseudo = (const float*)d_in[6];
    const float* wfc    = (const float*)d_in[7];
    const float* mu     = (const float*)d_in[8];
    const float* invs   = (const float*)d_in[9];
    const float* bias   = (const float*)d_in[10];
    float*       out    = (float*)d_out;
    float*       nodefeat = (float*)d_ws;   // [N, 256] fp32 = 51.2 MB scratch

    // Stage 1: 3125 mtiles * 4 ngroups = 12500 waves, 8 waves per 256-thread block
    const int gemm_waves  = MTILES * NGROUPS;
    const int gemm_blocks = (gemm_waves + 7) / 8;
    gmm_gemm_kernel<<<gemm_blocks, 256, 0, stream>>>(feat, wfc, nodefeat);

    // Stage 2: one wave per row
    const int edge_blocks = (NNODE + 7) / 8;
    gmm_edge_kernel<<<edge_blocks, 256, 0, stream>>>(rowptr, colind, pseudo, nodefeat,
                                                     mu, invs, bias, out);
}